// EmbeddingsInput_24240795418963
// MI455X (gfx1250) — compile-verified
//
#include <hip/hip_runtime.h>

typedef __attribute__((ext_vector_type(2))) float v2f;
typedef __attribute__((ext_vector_type(8))) float v8f;

#define B_ 64
#define N_ 2048
#define F_ 16
#define E_ 64
#define OUT_ 10
#define K_ (E_ * F_)          // 1024
#define NCH 128               // cells per chunk
#define NCHUNKS (N_ / NCH)    // 16
#define NSUB 4                // n-subsets per block

// Monotonic float -> uint mapping so unsigned atomic max == float max.
__device__ __forceinline__ unsigned fmap(float f) {
    unsigned u = __float_as_uint(f);
    return (u & 0x80000000u) ? ~u : (u | 0x80000000u);
}

// ---------------------------------------------------------------- init
__global__ void init_ws(unsigned* __restrict__ ws) {
    int i = blockIdx.x * blockDim.x + threadIdx.x;
    if (i < B_ * K_) ws[i] = 0x007FFFFFu;   // fmap(-inf)
}

// ----------------------------------------------- gather + outer + max
// grid: (NCHUNKS, B), block: 256 (8 waves).
// Thread t owns e = t&63 and ALL 16 f accumulators; subset s = t>>6 strides
// over n. Each embedding element is loaded exactly once per block
// (coalesced 128B per wave); features are LDS-broadcast. Subsets are
// pre-reduced in LDS, then one native u32 atomic-max per (e,f).
__global__ void __launch_bounds__(256)
pool_kernel(const int* __restrict__ ids,
            const float* __restrict__ features,
            const float* __restrict__ emb,
            unsigned* __restrict__ ws) {
    __shared__ int s_ids[NCH];
    __shared__ __align__(16) float s_feat[NCH * F_];   // 8 KB
    __shared__ float s_red[NSUB * K_];                 // 16 KB

    const int b  = blockIdx.y;
    const int n0 = blockIdx.x * NCH;
    const int t  = threadIdx.x;

    if (t < NCH) s_ids[t] = ids[b * N_ + n0 + t];
    const float* fbase = features + (size_t)(b * N_ + n0) * F_;
    for (int i = t; i < NCH * F_; i += 256) s_feat[i] = fbase[i];
    __syncthreads();

    const int e = t & (E_ - 1);   // lane-consecutive -> coalesced emb reads
    const int s = t >> 6;         // 0..3, uniform per wave -> LDS broadcast

    float acc[F_];
#pragma unroll
    for (int f = 0; f < F_; ++f) acc[f] = -INFINITY;

    for (int n = s; n < NCH; n += NSUB) {
        const float ev = emb[(size_t)s_ids[n] * E_ + e];
        float fv[F_];
        const float4* fr = (const float4*)&s_feat[n * F_];
        *(float4*)&fv[0]  = fr[0];
        *(float4*)&fv[4]  = fr[1];
        *(float4*)&fv[8]  = fr[2];
        *(float4*)&fv[12] = fr[3];
#pragma unroll
        for (int f = 0; f < F_; ++f) acc[f] = fmaxf(acc[f], ev * fv[f]);
    }

    // partial results -> LDS, reduce the 4 subsets, then atomic to L2
#pragma unroll
    for (int f = 0; f < F_; ++f) s_red[s * K_ + (e * F_ + f)] = acc[f];
    __syncthreads();

    unsigned* wb = ws + b * K_;
    for (int p = t; p < K_; p += 256) {
        float m = fmaxf(fmaxf(s_red[p],         s_red[K_ + p]),
                        fmaxf(s_red[2*K_ + p],  s_red[3*K_ + p]));
        atomicMax(wb + p, fmap(m));   // global_atomic_max_u32, deterministic
    }
}

// ------------------------------------------- one-shot unmap (uint->float)
// Keeps the WMMA inner loop free of per-element decode VALU.
__global__ void unmap_ws(unsigned* __restrict__ ws) {
    int i = blockIdx.x * blockDim.x + threadIdx.x;
    if (i < B_ * K_) {
        unsigned u = ws[i];
        ws[i] = (u & 0x80000000u) ? (u & 0x7fffffffu) : ~u;  // float bits
    }
}

// ------------------------------------------------ WMMA dense layer
// [64 x 1024] @ [1024 x 10] + bias, padded N=10 -> 16.
// grid: 1 block, 128 threads = 4 waves; wave w owns M rows [16w, 16w+16).
__global__ void __launch_bounds__(128)
gemm_kernel(const unsigned* __restrict__ ws,
            const float* __restrict__ fc_w,
            const float* __restrict__ fc_b,
            float* __restrict__ out) {
    __shared__ float sB[K_ * 16];   // 64 KB, fc_w zero-padded to 16 cols

    const int t = threadIdx.x;      // 0..127
    for (int i = t; i < K_ * 16; i += 128) {
        const int k = i >> 4, n = i & 15;
        sB[i] = (n < OUT_) ? fc_w[k * OUT_ + n] : 0.0f;
    }
    __syncthreads();

    const int lane = t & 31;
    const int w    = t >> 5;        // wave id 0..3
    const int m0   = w * 16;        // M tile base
    const int g    = lane >> 4;     // lane group 0/1 -> K half
    const int mn   = lane & 15;     // M row (for A) / N col (for B, D)

    const float* arow = (const float*)ws + (size_t)(m0 + mn) * K_;

    v8f acc = {};
#pragma unroll 8
    for (int kk = 0; kk < K_; kk += 4) {
        const int k0 = kk + 2 * g;      // this lane-group's K pair
        v2f a, bv;
        // A 16x4 f32 layout: VGPR i <- A[m][2g + i]
        a.x = arow[k0];
        a.y = arow[k0 + 1];
        // B 4x16 f32 layout: VGPR i <- B[2g + i][n]
        bv.x = sB[k0 * 16 + mn];
        bv.y = sB[(k0 + 1) * 16 + mn];
        acc = __builtin_amdgcn_wmma_f32_16x16x4_f32(
            /*neg_a=*/false, a, /*neg_b=*/false, bv,
            /*c_mod=*/(short)0, acc, /*reuse_a=*/false, /*reuse_b=*/false);
    }

    if (mn < OUT_) {
        const float bias = fc_b[mn];
#pragma unroll
        for (int r = 0; r < 8; ++r) {
            // C/D layout: VGPR r, lane -> M = r + 8g, N = lane&15
            const int m = m0 + r + 8 * g;
            out[m * OUT_ + mn] = acc[r] + bias;
        }
    }
}

// ---------------------------------------------------------------- host
extern "C" void kernel_launch(void* const* d_in, const int* in_sizes, int n_in,
                              void* d_out, int out_size, void* d_ws, size_t ws_size,
                              hipStream_t stream) {
    const int*   ids      = (const int*)d_in[0];
    const float* features = (const float*)d_in[1];
    const float* emb      = (const float*)d_in[2];
    const float* fc_w     = (const float*)d_in[3];
    const float* fc_b     = (const float*)d_in[4];
    float*       out      = (float*)d_out;
    unsigned*    ws       = (unsigned*)d_ws;   // B_*K_ uints = 256 KB

    init_ws<<<(B_ * K_ + 255) / 256, 256, 0, stream>>>(ws);
    pool_kernel<<<dim3(NCHUNKS, B_), 256, 0, stream>>>(ids, features, emb, ws);
    unmap_ws<<<(B_ * K_ + 255) / 256, 256, 0, stream>>>(ws);
    gemm_kernel<<<1, 128, 0, stream>>>(ws, fc_w, fc_b, out);
}